// DecompGrid_20624432955487
// MI455X (gfx1250) — compile-verified
//
#include <hip/hip_runtime.h>

// K-planes decomposed-grid sampler for MI455X (gfx1250, wave32).
// Tables (~140MB) are L2-resident (192MB L2): gathers use default temporal
// hint; streamed coords/output use non-temporal to avoid evicting tables.
// No matmul structure exists in this workload, so no WMMA is used — the
// kernel is gather-rate bound, optimized via paired B64 corner loads and an
// LDS transpose for fully coalesced output stores.

namespace {

constexpr int C        = 16;         // channels
constexpr int GD       = 128;        // grid3d D=H=W
constexpr int PS       = 256;        // plane H=W
constexpr int LL       = 64;         // line length
constexpr unsigned GRID_CH  = (unsigned)GD * GD * GD;  // 2,097,152 floats/channel
constexpr unsigned PLANE_CH = (unsigned)PS * PS;       // 65,536 floats/channel
constexpr int BLK      = 256;        // 8 waves (wave32)
constexpr int PAD      = 33;         // LDS row pad: odd -> conflict-free

typedef float f4v __attribute__((ext_vector_type(4)));

// 8-byte pair with 4-byte alignment: lets the backend emit global_load_b64
// at dword alignment (x0/x1 corners are adjacent along the fastest axis).
struct alignas(4) F2 { float lo, hi; };

__device__ __forceinline__ F2 ld2(const float* p) {
    return *reinterpret_cast<const F2*>(p);
}
__device__ __forceinline__ float lerp1(float w, F2 v) {
    return fmaf(w, v.hi - v.lo, v.lo);
}

__device__ __forceinline__ void axis_idx(float coord, int size,
                                         unsigned& i0, float& w) {
    // p in [0, size-1] -> identical to reference clamp semantics, with the
    // guarantee i1 = i0 + 1 stays in bounds (enables paired B64 loads).
    float p = (coord + 1.0f) * 0.5f * (float)(size - 1);
    int i = (int)floorf(p);
    i = min(max(i, 0), size - 2);
    i0 = (unsigned)i;
    w = p - (float)i;
}

__global__ __launch_bounds__(BLK) void kplanes_sample_kernel(
    const float* __restrict__ xin,     // (N,4)
    const float* __restrict__ grid3d,  // (16,128,128,128)
    const float* __restrict__ plane0,  // (16,256,256)  dims (y,z)
    const float* __restrict__ plane1,  // (16,256,256)  dims (x,z)
    const float* __restrict__ plane2,  // (16,256,256)  dims (x,y)
    const float* __restrict__ line0,   // (16,64)
    float* __restrict__ out,           // (N,32) row-major
    int N)
{
    __shared__ float lds[BLK * PAD];

    const int tid = (int)threadIdx.x;
    const int n   = (int)blockIdx.x * BLK + tid;
    const int nc  = (n < N) ? n : (N - 1);   // tail threads compute garbage,
                                             // masked at the store stage

    // Streamed once: non-temporal so coords don't pollute L2.
    const f4v cd = __builtin_nontemporal_load(
        reinterpret_cast<const f4v*>(xin) + nc);

    // ---- 3D grid index/weights ----
    unsigned gix, giy, giz;
    float wx, wy, wz;
    axis_idx(cd.x, GD, gix, wx);
    axis_idx(cd.y, GD, giy, wy);
    axis_idx(cd.z, GD, giz, wz);
    const unsigned goff = (giz * GD + giy) * GD + gix;

    // ---- plane indices/weights (u -> W axis, v -> H axis) ----
    unsigned pu0, pv0, pu1, pv1, pu2, pv2;
    float wu0, wv0, wu1, wv1, wu2, wv2;
    axis_idx(cd.y, PS, pu0, wu0);  axis_idx(cd.z, PS, pv0, wv0);  // plane0:(1,2)
    axis_idx(cd.x, PS, pu1, wu1);  axis_idx(cd.z, PS, pv1, wv1);  // plane1:(0,2)
    axis_idx(cd.x, PS, pu2, wu2);  axis_idx(cd.y, PS, pv2, wv2);  // plane2:(0,1)
    const unsigned p0off = pv0 * PS + pu0;
    const unsigned p1off = pv1 * PS + pu1;
    const unsigned p2off = pv2 * PS + pu2;

    // ---- line index/weight (note: different mapping, p * (L-1)) ----
    float pt = cd.w * (float)(LL - 1);
    int it = (int)floorf(pt);
    it = min(max(it, 0), LL - 2);
    const float wt = pt - (float)it;

    // Per-table running pointers; corner deltas fold into 24-bit IOFFSET.
    const float* g  = grid3d + goff;
    const float* q0 = plane0 + p0off;
    const float* q1 = plane1 + p1off;
    const float* q2 = plane2 + p2off;
    const float* lp = line0 + (unsigned)it;

    float* const myrow = &lds[tid * PAD];

#pragma unroll
    for (int c = 0; c < C; ++c) {
        // 4 paired-corner B64 gathers for the 3D grid
        F2 g00 = ld2(g);                 // (z0,y0,x0..x1)
        F2 g01 = ld2(g + GD);            // (z0,y1)
        F2 g10 = ld2(g + GD * GD);       // (z1,y0)
        F2 g11 = ld2(g + GD * GD + GD);  // (z1,y1)
        // 2 paired B64 gathers per plane
        F2 a0 = ld2(q0);  F2 a1 = ld2(q0 + PS);
        F2 b0 = ld2(q1);  F2 b1 = ld2(q1 + PS);
        F2 e0 = ld2(q2);  F2 e1 = ld2(q2 + PS);
        // 1 paired B64 for the line
        F2 lv = ld2(lp);

        // trilinear
        float x00 = lerp1(wx, g00), x01 = lerp1(wx, g01);
        float x10 = lerp1(wx, g10), x11 = lerp1(wx, g11);
        float y0v = fmaf(wy, x01 - x00, x00);
        float y1v = fmaf(wy, x11 - x10, x10);
        float tri = fmaf(wz, y1v - y0v, y0v);
        // bilinear x3
        float u0a = lerp1(wu0, a0), u0b = lerp1(wu0, a1);
        float bl0 = fmaf(wv0, u0b - u0a, u0a);
        float u1a = lerp1(wu1, b0), u1b = lerp1(wu1, b1);
        float bl1 = fmaf(wv1, u1b - u1a, u1a);
        float u2a = lerp1(wu2, e0), u2b = lerp1(wu2, e1);
        float bl2 = fmaf(wv2, u2b - u2a, u2a);

        float spatial = tri * bl0 * bl1 * bl2;
        float param   = lerp1(wt, lv);

        myrow[c]     = spatial;   // ds_store, lane stride 33: conflict-free
        myrow[C + c] = param;

        g  += GRID_CH;
        q0 += PLANE_CH;
        q1 += PLANE_CH;
        q2 += PLANE_CH;
        lp += LL;
    }

    __syncthreads();

    // LDS transpose -> fully coalesced 128B/wave non-temporal stores.
    // Local flat index p = k*BLK + tid; point = p>>5, component = p&31.
    // Within a wave at fixed k, point is constant and component == lane:
    // reads lds[P*33 + lane] (conflict-free), writes 32 consecutive floats.
    const int blockStart = (int)blockIdx.x * (BLK * 2 * C);
    const int total      = N * (2 * C);
#pragma unroll
    for (int k = 0; k < 2 * C; ++k) {
        int p = k * BLK + tid;
        int f = blockStart + p;
        if (f < total) {
            float v = lds[(p >> 5) * PAD + (p & 31)];
            __builtin_nontemporal_store(v, out + f);
        }
    }
}

}  // namespace

extern "C" void kernel_launch(void* const* d_in, const int* in_sizes, int n_in,
                              void* d_out, int out_size, void* d_ws, size_t ws_size,
                              hipStream_t stream) {
    const float* x      = (const float*)d_in[0];
    const float* grid3d = (const float*)d_in[1];
    const float* plane0 = (const float*)d_in[2];
    const float* plane1 = (const float*)d_in[3];
    const float* plane2 = (const float*)d_in[4];
    const float* line0  = (const float*)d_in[5];
    float* out = (float*)d_out;

    const int N = in_sizes[0] / 4;
    const int nblocks = (N + BLK - 1) / BLK;

    kplanes_sample_kernel<<<nblocks, BLK, 0, stream>>>(
        x, grid3d, plane0, plane1, plane2, line0, out, N);
}